// HopToTokenEncoder_73512660239065
// MI455X (gfx1250) — compile-verified
//
#include <hip/hip_runtime.h>
#include <stdint.h>

#define NNODES   16384
#define DFEAT    64
#define KHOPS    4
#define MAXDEG   128
#define OUTSTRIDE ((KHOPS + 1) * DFEAT)   // 320 floats per node row in d_out

// workspace layout (bytes)
#define BM_WORDS  ((size_t)NNODES * NNODES / 32)      // 8,388,608 words = 32 MB
#define BM_BYTES  (BM_WORDS * 4)
#define CNT_BYTES ((size_t)NNODES * 4)                // 64 KB
#define COL_BYTES ((size_t)NNODES * MAXDEG * 2)       // 4 MB (u16 columns)

typedef unsigned int  v4u __attribute__((ext_vector_type(4)));
typedef int           v8i __attribute__((ext_vector_type(8)));
typedef int           v4i __attribute__((ext_vector_type(4)));

// ---------------------------------------------------------------- clear ws
__global__ void clear_ws_kernel(unsigned* bm_and_cnt, unsigned nwords) {
    unsigned i = blockIdx.x * blockDim.x + threadIdx.x;
    unsigned stride = gridDim.x * blockDim.x;
    for (; i < nwords; i += stride) bm_and_cnt[i] = 0u;
}

// ------------------------------------------------- build dedup'd ELL adjacency
// adj[src,dst] = 1 (set semantics): bitmap atomicOr dedups duplicate edges.
__global__ void build_adj_kernel(const int* __restrict__ ei, int E,
                                 unsigned* __restrict__ bm,
                                 int* __restrict__ cnt,
                                 unsigned short* __restrict__ col) {
    int e = blockIdx.x * blockDim.x + threadIdx.x;
    if (e >= E) return;
    unsigned src = (unsigned)ei[e]     & (NNODES - 1);
    unsigned dst = (unsigned)ei[E + e] & (NNODES - 1);
    unsigned lin  = src * (unsigned)NNODES + dst;
    unsigned word = lin >> 5;
    unsigned bit  = 1u << (dst & 31u);
    unsigned old  = atomicOr(&bm[word], bit);
    if (!(old & bit)) {
        int slot = atomicAdd(&cnt[src], 1);
        if (slot < MAXDEG) col[src * MAXDEG + slot] = (unsigned short)dst;
    }
}

// ---------------------------------------------------------------- hop 0 copy
__global__ void copy_hop0_kernel(const float4* __restrict__ x4, float* __restrict__ out) {
    int i = blockIdx.x * blockDim.x + threadIdx.x;   // N*D/4 threads
    int node = i >> 4;         // 16 float4 per node row
    int f4   = i & 15;
    *(float4*)&out[(size_t)node * OUTSTRIDE + f4 * 4] = x4[i];
}

// ------------------------------------------- TDM gather-mode descriptor issue
// Gathers `nvalid` rows (64 f32 each) of a 2D tensor at `gaddr` with row
// stride `src_stride` (elements) into LDS at `lds_addr`, rows selected by
// 16-bit indices packed at colw[g*8 .. g*8+7].
__device__ __forceinline__ void issue_gather(unsigned long long gaddr, int src_stride,
                                             unsigned lds_addr,
                                             const unsigned* __restrict__ colw,
                                             int g, int nvalid, int lane) {
    unsigned myw = 0;
    if (lane < 8) myw = colw[g * 8 + lane];
    int i0 = __builtin_amdgcn_readlane((int)myw, 0);
    int i1 = __builtin_amdgcn_readlane((int)myw, 1);
    int i2 = __builtin_amdgcn_readlane((int)myw, 2);
    int i3 = __builtin_amdgcn_readlane((int)myw, 3);
    int i4 = __builtin_amdgcn_readlane((int)myw, 4);
    int i5 = __builtin_amdgcn_readlane((int)myw, 5);
    int i6 = __builtin_amdgcn_readlane((int)myw, 6);
    int i7 = __builtin_amdgcn_readlane((int)myw, 7);

    v4u g0;
    g0.x = 1u | (1u << 31);                      // count=1, gather_mode=1, 16-bit idx
    g0.y = lds_addr;                             // LDS destination
    g0.z = (unsigned)gaddr;                      // global_addr[31:0]
    g0.w = ((unsigned)(gaddr >> 32) & 0x01FFFFFFu) | (2u << 30);  // addr[56:32], type=2

    v8i g1;
    g1[0] = (int)(2u << 16);                     // data_size=2 -> 4 bytes
    g1[1] = (int)((unsigned)DFEAT << 16);        // tensor_dim0 = 64
    g1[2] = (int)((unsigned)NNODES << 16);       // tensor_dim1 = 16384
    g1[3] = (int)((unsigned)DFEAT << 16);        // tile_dim0 = 64
    g1[4] = nvalid & 0xFFFF;                     // tile_dim1 = #valid gather indices
    g1[5] = src_stride;                          // tensor_dim0_stride low 32 (elements)
    g1[6] = 0;
    g1[7] = 0;

    v4i g2 = {i0, i1, i2, i3};                   // row_index_0..7  (packed u16)
    v4i g3 = {i4, i5, i6, i7};                   // row_index_8..15 (packed u16)

#if defined(__clang_major__) && __clang_major__ >= 23
    v8i gpad = {0, 0, 0, 0, 0, 0, 0, 0};
    __builtin_amdgcn_tensor_load_to_lds(g0, g1, g2, g3, gpad, 0);
#else
    __builtin_amdgcn_tensor_load_to_lds(g0, g1, g2, g3, 0);
#endif
}

// ------------------------------------------------------------------- hop k
// One wave32 per node; double-buffered TDM gather pipelined against the
// LDS reduction (TENSORcnt is in-order per wave: after issuing buffer b+1,
// s_wait_tensorcnt 1 guarantees buffer b is complete).
__global__ __launch_bounds__(256) void hop_kernel(const float* __restrict__ src_base,
                                                  int src_stride,
                                                  const int* __restrict__ cnt,
                                                  const unsigned short* __restrict__ col,
                                                  float* __restrict__ out,
                                                  int hop) {
    __shared__ float stage[8][2][16][DFEAT];   // 8 waves * 2 bufs * 4KB = 64 KB

    const int lane = threadIdx.x & 31;
    const int wv   = threadIdx.x >> 5;

    int node = blockIdx.x * 8 + wv;
    node = __builtin_amdgcn_readfirstlane(node);

    int deg = cnt[node];
    deg = __builtin_amdgcn_readfirstlane(deg);
    if (deg > MAXDEG) deg = MAXDEG;
    const int ngroups = (deg + 15) >> 4;

    // LDS byte offsets of this wave's two staging buffers (low 32 bits of a
    // generic LDS pointer are the LDS offset).
    unsigned lds0 = (unsigned)(uintptr_t)&stage[wv][0][0][0];
    unsigned lds1 = (unsigned)(uintptr_t)&stage[wv][1][0][0];
    lds0 = (unsigned)__builtin_amdgcn_readfirstlane((int)lds0);
    lds1 = (unsigned)__builtin_amdgcn_readfirstlane((int)lds1);

    unsigned long long gaddr = (unsigned long long)(uintptr_t)src_base;
    const unsigned* colw = (const unsigned*)(col + (size_t)node * MAXDEG); // 8 dwords / 16 idx

    float acc0 = 0.0f, acc1 = 0.0f;

    if (ngroups > 0) {
        int nv0 = (deg < 16) ? deg : 16;
        issue_gather(gaddr, src_stride, lds0, colw, 0, nv0, lane);

        for (int g = 0; g < ngroups; ++g) {
            if (g + 1 < ngroups) {
                int nvn = deg - (g + 1) * 16;
                if (nvn > 16) nvn = 16;
                // prior reads of the alternate buffer must retire before the
                // DMA engine overwrites it
                asm volatile("s_wait_dscnt 0" ::: "memory");
                issue_gather(gaddr, src_stride, ((g + 1) & 1) ? lds1 : lds0,
                             colw, g + 1, nvn, lane);
                __builtin_amdgcn_s_wait_tensorcnt(1);   // group g complete
            } else {
                __builtin_amdgcn_s_wait_tensorcnt(0);   // last group complete
            }

            int nvalid = deg - g * 16;
            if (nvalid > 16) nvalid = 16;
            const float* buf = &stage[wv][g & 1][0][0];
            // each lane owns features 2*lane and 2*lane+1 -> ds_load_b64
            for (int r = 0; r < nvalid; ++r) {
                float2 v = *(const float2*)&buf[r * DFEAT + 2 * lane];
                acc0 += v.x;
                acc1 += v.y;
            }
        }
    }

    float* orow = out + (size_t)node * OUTSTRIDE + (size_t)hop * DFEAT;
    *(float2*)&orow[2 * lane] = make_float2(acc0, acc1);   // global_store_b64
}

// ------------------------------------------------------------------ launch
extern "C" void kernel_launch(void* const* d_in, const int* in_sizes, int n_in,
                              void* d_out, int out_size, void* d_ws, size_t ws_size,
                              hipStream_t stream) {
    const float* x  = (const float*)d_in[0];
    const int*   ei = (const int*)d_in[1];
    const int E = in_sizes[1] / 2;

    float* out = (float*)d_out;

    unsigned*       bm  = (unsigned*)d_ws;
    int*            cnt = (int*)((char*)d_ws + BM_BYTES);
    unsigned short* col = (unsigned short*)((char*)d_ws + BM_BYTES + CNT_BYTES);

    // 1) clear bitmap + degree counters (contiguous words)
    const unsigned clear_words = (unsigned)(BM_WORDS + NNODES);
    clear_ws_kernel<<<4096, 256, 0, stream>>>(bm, clear_words);

    // 2) build dedup'd ELL adjacency
    build_adj_kernel<<<(E + 255) / 256, 256, 0, stream>>>(ei, E, bm, cnt, col);

    // 3) hop 0 = x itself (b128 copy)
    copy_hop0_kernel<<<(NNODES * DFEAT / 4) / 256, 256, 0, stream>>>((const float4*)x, out);

    // 4) hops 1..4 (one wave per node, 8 waves per block)
    hop_kernel<<<NNODES / 8, 256, 0, stream>>>(x, DFEAT, cnt, col, out, 1);
    for (int k = 2; k <= KHOPS; ++k) {
        const float* prev = out + (size_t)(k - 1) * DFEAT;   // hop k-1 slice, stride 320
        hop_kernel<<<NNODES / 8, 256, 0, stream>>>(prev, OUTSTRIDE, cnt, col, out, k);
    }
}